// PhyGRUResidualCell_7129645711672
// MI455X (gfx1250) — compile-verified
//
#include <hip/hip_runtime.h>

typedef float v2f __attribute__((ext_vector_type(2)));
typedef float v8f __attribute__((ext_vector_type(8)));

#define DT_C 0.01f
#define BATCH 2097152
#define NTILES (BATCH / 16)
#define WAVES_PER_BLOCK 4
#define BLOCK_THREADS (WAVES_PER_BLOCK * 32)
#define NUM_BLOCKS 4096

__device__ __forceinline__ float fast_tanh(float x) {
#if __has_builtin(__builtin_amdgcn_tanhf)
    return __builtin_amdgcn_tanhf(x);
#else
    return tanhf(x);
#endif
}

__device__ __forceinline__ float fast_sigmoid(float x) {
    return 1.0f / (1.0f + __expf(-x));
}

__global__ __launch_bounds__(BLOCK_THREADS) void phygru_kernel(
    const float* __restrict__ state, const float* __restrict__ u,
    const float* __restrict__ A,     const float* __restrict__ Bm,
    const float* __restrict__ C,     const float* __restrict__ Wz,
    const float* __restrict__ bz,    const float* __restrict__ W1,
    const float* __restrict__ b1,    const float* __restrict__ W2,
    const float* __restrict__ b2,    const float* __restrict__ Wb1,
    const float* __restrict__ bb1,   const float* __restrict__ Wb2,
    const float* __restrict__ bb2,
    float* __restrict__ out_ns, float* __restrict__ out_y)
{
    // ---- block-shared constants ----
    __shared__ float cW2[256];   // 4x64
    __shared__ float cWb2[96];   // 3x32
    __shared__ float cA[9], cB[9], cC[9], cb2[4], cbb2[3];
    // ---- per-wave working tiles ----
    __shared__ float su   [WAVES_PER_BLOCK][16][12]; // [state|u|pad]  K padded to 12
    __shared__ float hbuf [WAVES_PER_BLOCK][16][64]; // tanh(su@W1^T + b1)
    __shared__ float zbuf [WAVES_PER_BLOCK][16][8];  // sigmoid gate (7 valid)
    __shared__ float hbbuf[WAVES_PER_BLOCK][16][96]; // tanh branch hidden (3x32)

    const int tid = threadIdx.x;
    for (int i = tid; i < 256; i += BLOCK_THREADS) cW2[i]  = W2[i];
    for (int i = tid; i < 96;  i += BLOCK_THREADS) cWb2[i] = Wb2[i];
    if (tid < 9) { cA[tid] = A[tid]; cB[tid] = Bm[tid]; cC[tid] = C[tid]; }
    if (tid < 4) cb2[tid] = b2[tid];
    if (tid < 3) cbb2[tid] = bb2[tid];
    __syncthreads();

    const int lane = tid & 31;
    const int wid  = tid >> 5;
    const int n16  = lane & 15;
    const int half = lane >> 4;

    // ---- loop-invariant WMMA B-fragments (lane holds column n16; K = c*4 + 2*half + {0,1}) ----
    v2f w1f[4][3];
    #pragma unroll
    for (int t = 0; t < 4; ++t)
        #pragma unroll
        for (int c = 0; c < 3; ++c) {
            const int n = t * 16 + n16;
            const int k0 = c * 4 + 2 * half;
            w1f[t][c].x = (k0     < 10) ? W1[n * 10 + k0]     : 0.0f;
            w1f[t][c].y = (k0 + 1 < 10) ? W1[n * 10 + k0 + 1] : 0.0f;
        }
    v2f wzf[3];
    #pragma unroll
    for (int c = 0; c < 3; ++c) {
        const int k0 = c * 4 + 2 * half;
        wzf[c].x = (n16 < 7 && k0     < 10) ? Wz[n16 * 10 + k0]     : 0.0f;
        wzf[c].y = (n16 < 7 && k0 + 1 < 10) ? Wz[n16 * 10 + k0 + 1] : 0.0f;
    }
    v2f wbf[6][2];
    float bb1v[6];
    #pragma unroll
    for (int t = 0; t < 6; ++t) {
        const int kh = t * 16 + n16;          // 0..95 -> (branch k, hidden h)
        const int kb = kh >> 5, hh = kh & 31;
        #pragma unroll
        for (int c = 0; c < 2; ++c) {
            const int k0 = c * 4 + 2 * half;
            wbf[t][c].x = (k0     < 6) ? Wb1[kb * 192 + hh * 6 + k0]     : 0.0f;
            wbf[t][c].y = (k0 + 1 < 6) ? Wb1[kb * 192 + hh * 6 + k0 + 1] : 0.0f;
        }
        bb1v[t] = bb1[kh];
    }
    float b1v[4];
    #pragma unroll
    for (int t = 0; t < 4; ++t) b1v[t] = b1[t * 16 + n16];
    const float bzv = (n16 < 7) ? bz[n16] : 0.0f;

    // zero the K-padding columns once (never overwritten inside the loop)
    if (half == 0) { su[wid][n16][10] = 0.0f; su[wid][n16][11] = 0.0f; }

    const int wavesTotal = gridDim.x * (BLOCK_THREADS >> 5);
    const int waveG = blockIdx.x * (BLOCK_THREADS >> 5) + wid;

    for (int tile = waveG; tile < NTILES; tile += wavesTotal) {
        const long rowBase = (long)tile * 16;
        const float* sp = state + rowBase * 7;
        const float* up = u + rowBase * 3;

        // ---- coalesced stage of 16x7 state + 16x3 u into su tile ----
        #pragma unroll
        for (int i = 0; i < 4; ++i) {
            const int idx = lane + 32 * i;
            if (idx < 112) su[wid][idx / 7][idx % 7] = sp[idx];
        }
        #pragma unroll
        for (int i = 0; i < 2; ++i) {
            const int idx = lane + 32 * i;
            if (idx < 48) su[wid][idx / 3][7 + idx % 3] = up[idx];
        }
        // prefetch next tile (global_prefetch_b8)
        {
            const long nt = (long)tile + wavesTotal;
            if (nt < NTILES) {
                __builtin_prefetch(state + nt * 112 + lane * 4, 0, 0);
                __builtin_prefetch(u + nt * 48 + lane * 2, 0, 0);
            }
        }
        __builtin_amdgcn_wave_barrier();

        // ---- A-fragments (16x4 fp32 layout: lane m=n16, VGPR0/1 = K 2*half / 2*half+1) ----
        v2f af[3];
        #pragma unroll
        for (int c = 0; c < 3; ++c) {
            const int k0 = c * 4 + 2 * half;
            af[c].x = su[wid][n16][k0];
            af[c].y = su[wid][n16][k0 + 1];
        }

        // ---- L1: su(16x12) @ W1^T(12x64) ; tanh ; -> LDS ----
        #pragma unroll
        for (int t = 0; t < 4; ++t) {
            v8f acc = {};
            #pragma unroll
            for (int c = 0; c < 3; ++c)
                acc = __builtin_amdgcn_wmma_f32_16x16x4_f32(
                    false, af[c], false, w1f[t][c], (short)0, acc, false, false);
            #pragma unroll
            for (int j = 0; j < 8; ++j) {
                const int row = j + 8 * half;
                hbuf[wid][row][t * 16 + n16] = fast_tanh(acc[j] + b1v[t]);
            }
        }

        // ---- gate: su @ Wz^T (7 valid cols) ; sigmoid ; -> LDS ----
        {
            v8f acc = {};
            #pragma unroll
            for (int c = 0; c < 3; ++c)
                acc = __builtin_amdgcn_wmma_f32_16x16x4_f32(
                    false, af[c], false, wzf[c], (short)0, acc, false, false);
            #pragma unroll
            for (int j = 0; j < 8; ++j) {
                const int row = j + 8 * half;
                if (n16 < 8) zbuf[wid][row][n16] = fast_sigmoid(acc[j] + bzv);
            }
        }

        // ---- branch layer: bin(16x8) @ Wb^T(8x96) ; tanh ; -> LDS ----
        v2f bf[2];
        #pragma unroll
        for (int c = 0; c < 2; ++c) {
            const int k0 = c * 4 + 2 * half;
            const int k1 = k0 + 1;
            bf[c].x = (k0 < 3) ? su[wid][n16][k0] : ((k0 < 6) ? su[wid][n16][k0 + 4] : 0.0f);
            bf[c].y = (k1 < 3) ? su[wid][n16][k1] : ((k1 < 6) ? su[wid][n16][k1 + 4] : 0.0f);
        }
        #pragma unroll
        for (int t = 0; t < 6; ++t) {
            v8f acc = {};
            #pragma unroll
            for (int c = 0; c < 2; ++c)
                acc = __builtin_amdgcn_wmma_f32_16x16x4_f32(
                    false, bf[c], false, wbf[t][c], (short)0, acc, false, false);
            #pragma unroll
            for (int j = 0; j < 8; ++j) {
                const int row = j + 8 * half;
                hbbuf[wid][row][t * 16 + n16] = fast_tanh(acc[j] + bb1v[t]);
            }
        }
        __builtin_amdgcn_wave_barrier();

        // ---- per-row epilogue: lanes 0-15 state path, lanes 16-31 output path ----
        const int r = n16;
        float st[10];
        #pragma unroll
        for (int i = 0; i < 10; ++i) st[i] = su[wid][r][i];

        if (half == 0) {
            // physics: s_next = s + DT*(A s + Bm u)
            float sn[3];
            #pragma unroll
            for (int i = 0; i < 3; ++i) {
                float d = 0.0f;
                #pragma unroll
                for (int j = 0; j < 3; ++j)
                    d += cA[i * 3 + j] * st[j] + cB[i * 3 + j] * st[7 + j];
                sn[i] = st[i] + DT_C * d;
            }
            // latent: latent + DT*(h @ W2^T + b2)
            float d0 = cb2[0], d1 = cb2[1], d2 = cb2[2], d3 = cb2[3];
            #pragma unroll 8
            for (int h = 0; h < 64; ++h) {
                const float hv = hbuf[wid][r][h];
                d0 += cW2[h] * hv;       d1 += cW2[64 + h] * hv;
                d2 += cW2[128 + h] * hv; d3 += cW2[192 + h] * hv;
            }
            float cand[7];
            cand[0] = sn[0]; cand[1] = sn[1]; cand[2] = sn[2];
            cand[3] = st[3] + DT_C * d0; cand[4] = st[4] + DT_C * d1;
            cand[5] = st[5] + DT_C * d2; cand[6] = st[6] + DT_C * d3;
            #pragma unroll
            for (int i = 0; i < 7; ++i) {
                const float zi = zbuf[wid][r][i];
                su[wid][r][i] = st[i] + zi * (cand[i] - st[i]);  // z*cand + (1-z)*state
            }
        } else {
            // y = C s + (Wb2 . tanh_branch + bb2)
            #pragma unroll
            for (int i = 0; i < 3; ++i) {
                float yp = cC[i * 3] * st[0] + cC[i * 3 + 1] * st[1] + cC[i * 3 + 2] * st[2];
                float resid = cbb2[i];
                #pragma unroll 8
                for (int h = 0; h < 32; ++h)
                    resid += cWb2[i * 32 + h] * hbbuf[wid][r][i * 32 + h];
                su[wid][r][7 + i] = yp + resid;
            }
        }
        __builtin_amdgcn_wave_barrier();

        // ---- coalesced flush of outputs ----
        float* onp = out_ns + rowBase * 7;
        #pragma unroll
        for (int i = 0; i < 4; ++i) {
            const int idx = lane + 32 * i;
            if (idx < 112) onp[idx] = su[wid][idx / 7][idx % 7];
        }
        float* oyp = out_y + rowBase * 3;
        #pragma unroll
        for (int i = 0; i < 2; ++i) {
            const int idx = lane + 32 * i;
            if (idx < 48) oyp[idx] = su[wid][idx / 3][7 + idx % 3];
        }
    }
}

extern "C" void kernel_launch(void* const* d_in, const int* in_sizes, int n_in,
                              void* d_out, int out_size, void* d_ws, size_t ws_size,
                              hipStream_t stream) {
    (void)in_sizes; (void)n_in; (void)out_size; (void)d_ws; (void)ws_size;
    const float* state = (const float*)d_in[0];
    const float* u     = (const float*)d_in[1];
    const float* A     = (const float*)d_in[2];
    const float* Bm    = (const float*)d_in[3];
    const float* C     = (const float*)d_in[4];
    const float* Wz    = (const float*)d_in[5];
    const float* bz    = (const float*)d_in[6];
    const float* W1    = (const float*)d_in[7];
    const float* b1    = (const float*)d_in[8];
    const float* W2    = (const float*)d_in[9];
    const float* b2    = (const float*)d_in[10];
    const float* Wb1   = (const float*)d_in[11];
    const float* bb1   = (const float*)d_in[12];
    const float* Wb2   = (const float*)d_in[13];
    const float* bb2   = (const float*)d_in[14];
    float* out_ns = (float*)d_out;                    // B x 7
    float* out_y  = (float*)d_out + (long)BATCH * 7;  // B x 3

    phygru_kernel<<<NUM_BLOCKS, BLOCK_THREADS, 0, stream>>>(
        state, u, A, Bm, C, Wz, bz, W1, b1, W2, b2, Wb1, bb1, Wb2, bb2,
        out_ns, out_y);
}